// WeightedHausdorffDistance_36412732735861
// MI455X (gfx1250) — compile-verified
//
#include <hip/hip_runtime.h>
#include <math.h>

#define HH 256
#define WW 256
#define PIX (HH * WW)          // 65536 pixels per image
#define BATCH 8
#define NGT 256
#define EPSV 1e-6f
#define MAXD 362.03867196751236f   // sqrt(256^2 + 256^2)

typedef __attribute__((ext_vector_type(2))) float v2f;
typedef __attribute__((ext_vector_type(8))) float v8f;

// ---------------------------------------------------------------------------
// Zero the accumulator workspace (harness poisons it with 0xAA only once).
// ---------------------------------------------------------------------------
__global__ void whd_zero(float* __restrict__ ws, int n) {
    int i = blockIdx.x * blockDim.x + threadIdx.x;
    if (i < n) ws[i] = 0.0f;
}

// ---------------------------------------------------------------------------
// Main kernel: one wave = 16 pixels x all 256 GT points of one image.
// Block = 256 threads = 8 waves sharing one image; per-GT column sums are
// combined in LDS before one global atomic per (block, GT point).
// d^2 tiles come from V_WMMA_F32_16X16X4_F32 with
//   A row i  = [y, x, y^2+x^2, 1]
//   B col j  = [-2*gy, -2*gx, 1, gy^2+gx^2]
// so A*B = d^2 exactly (rank-4 expansion of the squared distance).
// Clamps/mins are done on float bit patterns (v_max_i32 / v_min_u32) to
// avoid clang's minnum/maxnum canonicalization ops in the hot loop.
// ---------------------------------------------------------------------------
__global__ __launch_bounds__(256) void whd_main(const float* __restrict__ prob,
                                                const float* __restrict__ gt,
                                                float* __restrict__ nest,   // [BATCH]
                                                float* __restrict__ t1num,  // [BATCH]
                                                float* __restrict__ gsum)   // [BATCH*NGT]
{
    __shared__ float gsum_l[NGT];
    const int tid = threadIdx.x;
    for (int t = tid; t < NGT; t += 256) gsum_l[t] = 0.0f;
    __syncthreads();

    const int wave = tid >> 5;
    const int lane = tid & 31;
    const int m    = lane & 15;     // column within 16-wide tile
    const int hi   = lane >> 4;     // which 16-lane half

    const int b    = blockIdx.x >> 9;                     // image (512 blocks/img)
    const int tile = ((blockIdx.x & 511) << 3) + wave;    // 0..4095 pixel tile
    const int i0   = tile << 4;                           // base pixel index

    // ---- A matrix (16x4 f32): lanes 0-15 hold K=0,1 ; lanes 16-31 hold K=2,3
    const int   ip = i0 + m;
    const float y  = (float)(ip >> 8);
    const float x  = (float)(ip & 255);
    const float n2 = y * y + x * x;
    v2f a;
    a.x = hi ? n2   : y;    // K=0 or K=2
    a.y = hi ? 1.0f : x;    // K=1 or K=3

    // ---- p values for the 8 rows this lane owns in D: M = k + 8*hi
    float pk[8], base[8];
    const float* pbase = prob + b * PIX + i0 + (hi << 3);
#pragma unroll
    for (int k = 0; k < 8; ++k) {
        pk[k]   = pbase[k];
        base[k] = (1.0f - pk[k]) * MAXD + EPSV;   // w = fma(p, d, base)
    }

    // running min of d, kept as unsigned bit pattern (d >= 0 always)
    unsigned minu[8];
#pragma unroll
    for (int k = 0; k < 8; ++k) minu[k] = 0x7f7fffffu;    // bits(FLT_MAX)

    // ---- preload all 16 B-matrix tiles (per-lane column j = jt*16 + m)
    const float* gtb = gt + b * NGT * 2;
    float bx[16], by[16];
#pragma unroll
    for (int jt = 0; jt < 16; ++jt) {
        const int   j  = (jt << 4) + m;
        const float gy = gtb[2 * j + 0];
        const float gx = gtb[2 * j + 1];
        bx[jt] = hi ? 1.0f                 : (-2.0f * gy);   // K=0 / K=2 row
        by[jt] = hi ? (gy * gy + gx * gx)  : (-2.0f * gx);   // K=1 / K=3 row
    }

    for (int jt = 0; jt < 16; ++jt) {
        v2f bm;
        bm.x = bx[jt];
        bm.y = by[jt];

        v8f c = {0.f, 0.f, 0.f, 0.f, 0.f, 0.f, 0.f, 0.f};
        v8f d2 = __builtin_amdgcn_wmma_f32_16x16x4_f32(
            false, a, false, bm, (short)0, c, false, false);

        float colacc = 0.0f;
#pragma unroll
        for (int k = 0; k < 8; ++k) {
            // clamp tiny negative rounding noise to +0.0 via signed-int max
            int d2i = __float_as_int(d2[k]);
            d2i = (d2i < 0) ? 0 : d2i;                            // v_max_i32
            const float dd = __builtin_amdgcn_sqrtf(__int_as_float(d2i));
            const unsigned ddu = __float_as_uint(dd);             // dd >= 0
            minu[k] = (ddu < minu[k]) ? ddu : minu[k];            // v_min_u32
            const float w  = __builtin_fmaf(pk[k], dd, base[k]);  // (1-p)*MAXD + p*d + eps
            const float w2 = w * w;
            const float w4 = w2 * w2;
            const float w8 = w4 * w4;
            colacc += __builtin_amdgcn_rcpf(w8 * w);              // w^-9
        }
        atomicAdd(&gsum_l[(jt << 4) + m], colacc);
    }

    // ---- min over all 256 GT points: reduce across each 16-lane half
    //      (integer-domain min on bit patterns, valid for non-negative floats)
#pragma unroll
    for (int k = 0; k < 8; ++k) {
        unsigned v = minu[k];
        unsigned o;
        o = (unsigned)__shfl_xor((int)v, 1, 32); v = (o < v) ? o : v;
        o = (unsigned)__shfl_xor((int)v, 2, 32); v = (o < v) ? o : v;
        o = (unsigned)__shfl_xor((int)v, 4, 32); v = (o < v) ? o : v;
        o = (unsigned)__shfl_xor((int)v, 8, 32); v = (o < v) ? o : v;
        minu[k] = v;
    }
    float t1 = 0.0f, sp = 0.0f;
#pragma unroll
    for (int k = 0; k < 8; ++k) {
        t1 += pk[k] * __uint_as_float(minu[k]);
        sp += pk[k];
    }
    if (m == 0) {                       // lane 0 -> rows 0-7, lane 16 -> rows 8-15
        atomicAdd(&t1num[b], t1);
        atomicAdd(&nest[b], sp);
    }

    __syncthreads();
    for (int t = tid; t < NGT; t += 256)
        atomicAdd(&gsum[b * NGT + t], gsum_l[t]);
}

// ---------------------------------------------------------------------------
// Finalize: gmean = (sum/65536)^(1/ALPHA), batch means, term1 division.
// ---------------------------------------------------------------------------
__global__ __launch_bounds__(256) void whd_finalize(const float* __restrict__ nest,
                                                    const float* __restrict__ t1num,
                                                    const float* __restrict__ gsum,
                                                    float* __restrict__ out)
{
    __shared__ float red[256];
    const int t = threadIdx.x;
    float s = 0.0f;
    for (int b = 0; b < BATCH; ++b) {
        const float mv = gsum[b * NGT + t] * (1.0f / (float)PIX);
        s += powf(mv, -1.0f / 9.0f);
    }
    red[t] = s;
    __syncthreads();
    for (int off = 128; off > 0; off >>= 1) {
        if (t < off) red[t] += red[t + off];
        __syncthreads();
    }
    if (t == 0) {
        const float term2 = red[0] / (float)(BATCH * NGT);   // mean_b mean_j
        float term1 = 0.0f;
        for (int b = 0; b < BATCH; ++b) term1 += t1num[b] / (nest[b] + EPSV);
        out[0] = term1 * (1.0f / (float)BATCH) + term2;
    }
}

// ---------------------------------------------------------------------------
extern "C" void kernel_launch(void* const* d_in, const int* in_sizes, int n_in,
                              void* d_out, int out_size, void* d_ws, size_t ws_size,
                              hipStream_t stream)
{
    const float* prob = (const float*)d_in[0];   // (8,1,256,256) f32
    const float* gt   = (const float*)d_in[1];   // (8,256,2)     f32

    float* ws    = (float*)d_ws;
    float* nest  = ws;          // 8
    float* t1num = ws + 8;      // 8
    float* gsum  = ws + 16;     // 8*256

    const int nz = 16 + BATCH * NGT;
    whd_zero<<<(nz + 255) / 256, 256, 0, stream>>>(ws, nz);
    whd_main<<<4096, 256, 0, stream>>>(prob, gt, nest, t1num, gsum);
    whd_finalize<<<1, 256, 0, stream>>>(nest, t1num, gsum, (float*)d_out);
}